// CostVolume3D_59081570125098
// MI455X (gfx1250) — compile-verified
//
#include <hip/hip_runtime.h>
#include <cstdint>

// Problem constants from the reference: (n, c, h, w) = (2, 32, 128, 256), MAX_DISP = 48
#define N_ 2
#define C_ 32
#define H_ 128
#define W_ 256
#define D_ 48

typedef float v4f __attribute__((ext_vector_type(4)));

// One block per (n, c, h) row-pair. Stage both 1KB rows in LDS via gfx1250
// async global->LDS b128 copies, then stream 48 shifted-difference rows out
// with non-temporal 128-bit stores (output doesn't fit L2; inputs do).
__global__ __launch_bounds__(256) void CostVolume3D_kernel(
    const float* __restrict__ l_fmap,
    const float* __restrict__ r_fmap,
    float* __restrict__ out)
{
    __shared__ float s_rows[2 * W_];   // [0..255] = l row, [256..511] = r row

    const int tid = threadIdx.x;
    const int bid = blockIdx.x;        // 0 .. N_*C_*H_ - 1
    const int h   = bid % H_;
    const int nc  = bid / H_;          // flattened (n*C + c), 0..63

    const size_t row_off = (size_t)(nc * H_ + h) * W_;  // element offset of this row

    // ---- Async-stage l/r rows into LDS (ASYNCcnt path) -------------------
    // Waves 0..3 (threads 0..127): 64 lanes copy the l row, 64 lanes the r
    // row, 16 bytes per lane.
    if (tid < 128) {
        const int    is_r  = tid >> 6;       // 0: l row, 1: r row
        const int    chunk = tid & 63;       // 16-byte chunk within the row
        const float* src   = (is_r ? r_fmap : l_fmap) + row_off + chunk * 4;
        const uint32_t lds_addr =
            (uint32_t)(uintptr_t)&s_rows[is_r * W_ + chunk * 4];
        asm volatile("global_load_async_to_lds_b128 %0, %1, off"
                     :
                     : "v"(lds_addr), "v"((uint64_t)(uintptr_t)src)
                     : "memory");
    }
    // Waves that issued async copies drain ASYNCcnt; others see it as 0.
    asm volatile("s_wait_asynccnt 0x0" ::: "memory");
    __syncthreads();

    // ---- Emit the 48 x 256 cost slab for this (n, c, h) ------------------
    // Fixed per-thread: 4 consecutive w's. 64 lanes (2 waves) cover one
    // contiguous 1KB output row; the 4 d-values per iteration come from the
    // upper 2 bits of tid.
    const int w4   = (tid & 63) * 4;
    const int dsub = tid >> 6;             // 0..3

    // Left values are loop-invariant for this thread: keep in registers.
    const float l0 = s_rows[w4 + 0];
    const float l1 = s_rows[w4 + 1];
    const float l2 = s_rows[w4 + 2];
    const float l3 = s_rows[w4 + 3];
    const float* s_r = &s_rows[W_];

    // Output layout (n, c, d, h, w): idx = ((nc*D + d)*H + h)*W + w
    float* out_base = out + ((size_t)nc * D_ * H_ + (size_t)h) * W_;

    #pragma unroll
    for (int db = 0; db < D_; db += 4) {
        const int d  = db + dsub;
        const int i0 = w4 + 0 - d;         // >= -47: still inside s_rows even
        const int i1 = w4 + 1 - d;         // when negative, so unconditional
        const int i2 = w4 + 2 - d;         // ds_load + v_cndmask is safe.
        const int i3 = w4 + 3 - d;
        const float r0 = s_r[i0];
        const float r1 = s_r[i1];
        const float r2 = s_r[i2];
        const float r3 = s_r[i3];
        v4f v;
        v.x = (i0 >= 0) ? (l0 - r0) : 1.0f;
        v.y = (i1 >= 0) ? (l1 - r1) : 1.0f;
        v.z = (i2 >= 0) ? (l2 - r2) : 1.0f;
        v.w = (i3 >= 0) ? (l3 - r3) : 1.0f;
        // Non-temporal b128 store: don't let the 403MB output stream evict
        // the L2-resident 16MB inputs.
        __builtin_nontemporal_store(
            v, (v4f*)(out_base + (size_t)d * (H_ * W_) + w4));
    }
}

extern "C" void kernel_launch(void* const* d_in, const int* in_sizes, int n_in,
                              void* d_out, int out_size, void* d_ws, size_t ws_size,
                              hipStream_t stream) {
    const float* l_fmap = (const float*)d_in[0];
    const float* r_fmap = (const float*)d_in[1];
    float* out = (float*)d_out;

    const int blocks = N_ * C_ * H_;   // 8192 blocks, 256 threads (8 waves) each
    CostVolume3D_kernel<<<blocks, 256, 0, stream>>>(l_fmap, r_fmap, out);
}